// SuperpixelGCN_5205500363108
// MI455X (gfx1250) — compile-verified
//
#include <hip/hip_runtime.h>
#include <hip/hip_bf16.h>

// ---------------------------------------------------------------------------
// GINE GNN forward for MI455X (gfx1250, wave32).
//  - edge message + scatter-add: per-edge f32 atomics (L2-resident agg)
//  - node update GEMMs (K=64) and classifier GEMM (K=128): V_WMMA_F32_16X16X4_F32
//  - segment mean/max pooling via atomic add + IEEE bit-trick atomic max
// ---------------------------------------------------------------------------

typedef float v2f __attribute__((ext_vector_type(2)));
typedef float v8f __attribute__((ext_vector_type(8)));

#define BN_EPS 1e-5f

// ---------------------------------------------------------------------------
// Edge kernel: emb = ea @ We + be ; msg = relu(h[src] + emb) ; agg[dst] += msg
// One thread per (edge, 4-feature chunk).
// ---------------------------------------------------------------------------
template<int D>
__global__ void gine_scatter(const float* __restrict__ h,
                             const int*   __restrict__ ei,   // [2*E]
                             const float* __restrict__ ea,   // [E,2]
                             const float* __restrict__ We,   // [2,D]
                             const float* __restrict__ be,   // [D]
                             float* __restrict__ agg,        // [N,D] zeroed
                             int E) {
  constexpr int CH = D / 4;
  int tid = blockIdx.x * blockDim.x + threadIdx.x;
  if (tid >= E * CH) return;
  int e  = tid / CH;
  int c4 = (tid - e * CH) * 4;
  int s = ei[e];
  int d = ei[E + e];
  float e0 = ea[2 * e], e1 = ea[2 * e + 1];
  const float* hs = h + (size_t)s * D + c4;
  float* ad = agg + (size_t)d * D + c4;
#pragma unroll
  for (int j = 0; j < 4; ++j) {
    float emb = fmaf(e0, We[c4 + j], fmaf(e1, We[D + c4 + j], be[c4 + j]));
    float m = fmaxf(hs[j] + emb, 0.0f);
    atomicAdd(&ad[j], m);
  }
}

// ---------------------------------------------------------------------------
// Layer 1 (K=12, two small GEMMs folded): h = relu(BN((x+agg)@W1+b1)) + x@Wr+br
// ---------------------------------------------------------------------------
__global__ void layer1_update(const float* __restrict__ x, const float* __restrict__ agg,
                              const float* __restrict__ W1, const float* __restrict__ b1,
                              const float* __restrict__ g1, const float* __restrict__ bt1,
                              const float* __restrict__ rm1, const float* __restrict__ rv1,
                              const float* __restrict__ Wr, const float* __restrict__ br,
                              float* __restrict__ h, int N) {
  int tid = blockIdx.x * blockDim.x + threadIdx.x;
  if (tid >= N * 64) return;
  int n = tid >> 6, f = tid & 63;
  const float* xr = x + (size_t)n * 12;
  const float* ar = agg + (size_t)n * 12;
  float s = 0.0f, r = 0.0f;
#pragma unroll
  for (int k = 0; k < 12; ++k) {
    float xv = xr[k];
    s = fmaf(xv + ar[k], W1[k * 64 + f], s);
    r = fmaf(xv, Wr[k * 64 + f], r);
  }
  float sc = g1[f] * rsqrtf(rv1[f] + BN_EPS);
  float v = sc * (s + b1[f] - rm1[f]) + bt1[f];
  h[tid] = fmaxf(v, 0.0f) + r + br[f];
}

// ---------------------------------------------------------------------------
// WMMA f32 GEMM: out[M x 64] = relu(maybeBN(A @ W + bias)) [+ res]
//   A = A1 (+ A2 if HAS_AGG), A is [M x KDIM] row-major, W is [KDIM x 64].
// Block = NWAVES waves, each wave computes a 16x64 tile via 16x16x4 f32 WMMA.
// ISA layouts (CDNA5 ISA §7.12.2):
//   A frag (16x4): lanes 0-15 -> K{k0,k0+1}, lanes 16-31 -> K{k0+2,k0+3}
//   B frag (4x16): V0 lanes0-15 K=k0 / lanes16-31 K=k0+2; V1: K=k0+1 / k0+3
//   C/D: VGPR j -> row j (lanes 0-15) / row j+8 (lanes 16-31), col = lane%16
// ---------------------------------------------------------------------------
template<int KDIM, int NWAVES, bool HAS_AGG, bool HAS_BN, bool HAS_RES>
__global__ __launch_bounds__(NWAVES * 32)
void gemm_n64_wmma(const float* __restrict__ A1, const float* __restrict__ A2,
                   const float* __restrict__ W,  const float* __restrict__ bias,
                   const float* __restrict__ g,  const float* __restrict__ bt,
                   const float* __restrict__ rm, const float* __restrict__ rv,
                   const float* __restrict__ res, float* __restrict__ out, int M) {
  static_assert(KDIM % 4 == 0, "K must be multiple of 4");
  constexpr int ROWS = NWAVES * 16;
  constexpr int LDA  = KDIM + 1;   // pad to dodge LDS bank conflicts
  constexpr int LDW  = 65;
  __shared__ float sA[ROWS * LDA];
  __shared__ float sW[KDIM * LDW];

  const int tid  = threadIdx.x;
  const int wave = tid >> 5;
  const int lane = tid & 31;
  const int half = lane >> 4;
  const int l16  = lane & 15;
  const int rowBlock = blockIdx.x * ROWS;

  // Stage W [KDIM x 64] into LDS (coalesced)
  for (int i = tid; i < KDIM * 64; i += NWAVES * 32)
    sW[(i >> 6) * LDW + (i & 63)] = W[i];
  // Stage A tile (+agg) into LDS
  for (int i = tid; i < ROWS * KDIM; i += NWAVES * 32) {
    int r = i / KDIM, c = i - r * KDIM;
    int gr = rowBlock + r;
    float a = 0.0f;
    if (gr < M) {
      a = A1[(size_t)gr * KDIM + c];
      if (HAS_AGG) a += A2[(size_t)gr * KDIM + c];
    }
    sA[r * LDA + c] = a;
  }
  __syncthreads();

  const int rbase = wave * 16;
  v8f acc[4] = {};
#pragma unroll
  for (int k0 = 0; k0 < KDIM; k0 += 4) {
    const int ka = k0 + 2 * half;
    v2f afrag;
    afrag.x = sA[(rbase + l16) * LDA + ka];
    afrag.y = sA[(rbase + l16) * LDA + ka + 1];
    const float* wp0 = &sW[ka * LDW];
    const float* wp1 = &sW[(ka + 1) * LDW];
#pragma unroll
    for (int t = 0; t < 4; ++t) {
      v2f bfrag;
      bfrag.x = wp0[t * 16 + l16];
      bfrag.y = wp1[t * 16 + l16];
      acc[t] = __builtin_amdgcn_wmma_f32_16x16x4_f32(
          /*neg_a=*/false, afrag, /*neg_b=*/false, bfrag,
          /*c_mod=*/(short)0, acc[t], /*reuse_a=*/false, /*reuse_b=*/false);
    }
  }

  // Epilogue: bias (+BN) + relu (+residual), scatter-store per C-layout.
#pragma unroll
  for (int t = 0; t < 4; ++t) {
    const int col = t * 16 + l16;
    const float bia = bias[col];
    float sc = 1.0f, sh = bia;
    if (HAS_BN) {
      sc = g[col] * rsqrtf(rv[col] + BN_EPS);
      sh = sc * (bia - rm[col]) + bt[col];
    }
#pragma unroll
    for (int j = 0; j < 8; ++j) {
      const int row = rowBlock + rbase + half * 8 + j;
      if (row < M) {
        float v = acc[t][j];
        v = HAS_BN ? fmaf(sc, v, sh) : (v + sh);
        v = fmaxf(v, 0.0f);
        if (HAS_RES) v += res[(size_t)row * 64 + col];
        out[(size_t)row * 64 + col] = v;
      }
    }
  }
}

// ---------------------------------------------------------------------------
// Pooling
// ---------------------------------------------------------------------------
__global__ void pool_init(float* __restrict__ sum, unsigned* __restrict__ mx,
                          int* __restrict__ counts, int G) {
  int tid = blockIdx.x * blockDim.x + threadIdx.x;
  if (tid < G * 64) { sum[tid] = 0.0f; mx[tid] = 0xFF800000u; }  // -inf
  if (tid < G) counts[tid] = 0;
}

__global__ void pool_scatter(const float* __restrict__ h, const int* __restrict__ batch,
                             float* __restrict__ sum, unsigned* __restrict__ mx,
                             int* __restrict__ counts, int N) {
  int tid = blockIdx.x * blockDim.x + threadIdx.x;
  if (tid >= N * 64) return;
  int n = tid >> 6, f = tid & 63;
  int b = batch[n];
  float v = h[tid];
  atomicAdd(&sum[b * 64 + f], v);
  unsigned* addr = &mx[b * 64 + f];
  if (v >= 0.0f) atomicMax((int*)addr, __float_as_int(v));
  else           atomicMin(addr, __float_as_uint(v));
  if (f == 0) atomicAdd(&counts[b], 1);
}

__global__ void pool_finalize(const float* __restrict__ sum, const unsigned* __restrict__ mx,
                              const int* __restrict__ counts, float* __restrict__ pooled, int G) {
  int tid = blockIdx.x * blockDim.x + threadIdx.x;
  if (tid >= G * 64) return;
  int b = tid >> 6, f = tid & 63;
  float cnt = (float)counts[b];
  pooled[b * 128 + f] = sum[tid] / fmaxf(cnt, 1.0f);
  float m = __uint_as_float(mx[tid]);
  pooled[b * 128 + 64 + f] = (cnt > 0.0f) ? m : 0.0f;
}

// Final projection: out[G x 10] = hc[G x 64] @ Wc2[64 x 10] + bc2
__global__ void cls_final(const float* __restrict__ hc, const float* __restrict__ Wc2,
                          const float* __restrict__ bc2, float* __restrict__ out, int G) {
  int tid = blockIdx.x * blockDim.x + threadIdx.x;
  if (tid >= G * 10) return;
  int gi = tid / 10, c = tid - gi * 10;
  float s = bc2[c];
#pragma unroll
  for (int k = 0; k < 64; ++k) s = fmaf(hc[gi * 64 + k], Wc2[k * 10 + c], s);
  out[tid] = s;
}

// ---------------------------------------------------------------------------
// Host launch
// ---------------------------------------------------------------------------
static inline int cdiv(long long a, long long b) { return (int)((a + b - 1) / b); }

extern "C" void kernel_launch(void* const* d_in, const int* in_sizes, int n_in,
                              void* d_out, int out_size, void* d_ws, size_t ws_size,
                              hipStream_t stream) {
  const float* x     = (const float*)d_in[0];
  const int*   ei    = (const int*)  d_in[1];
  const float* ea    = (const float*)d_in[2];
  const int*   batch = (const int*)  d_in[3];
  const float* We1 = (const float*)d_in[4],  *be1 = (const float*)d_in[5];
  const float* We2 = (const float*)d_in[6],  *be2 = (const float*)d_in[7];
  const float* We3 = (const float*)d_in[8],  *be3 = (const float*)d_in[9];
  const float* W1  = (const float*)d_in[10], *b1  = (const float*)d_in[11];
  const float* g1  = (const float*)d_in[12], *bt1 = (const float*)d_in[13];
  const float* rm1 = (const float*)d_in[14], *rv1 = (const float*)d_in[15];
  const float* W2  = (const float*)d_in[16], *b2  = (const float*)d_in[17];
  const float* g2  = (const float*)d_in[18], *bt2 = (const float*)d_in[19];
  const float* rm2 = (const float*)d_in[20], *rv2 = (const float*)d_in[21];
  const float* W3  = (const float*)d_in[22], *b3  = (const float*)d_in[23];
  const float* g3  = (const float*)d_in[24], *bt3 = (const float*)d_in[25];
  const float* rm3 = (const float*)d_in[26], *rv3 = (const float*)d_in[27];
  const float* Wr  = (const float*)d_in[28], *br  = (const float*)d_in[29];
  const float* Wc1 = (const float*)d_in[30], *bc1 = (const float*)d_in[31];
  const float* Wc2 = (const float*)d_in[32], *bc2 = (const float*)d_in[33];

  const int N = in_sizes[0] / 12;
  const int E = in_sizes[1] / 2;
  const int G = 512;

  // Workspace layout
  float* agg = (float*)d_ws;                    // N*64
  float* hA  = agg + (size_t)N * 64;            // N*64
  float* hB  = hA  + (size_t)N * 64;            // N*64
  float* sum = hB  + (size_t)N * 64;            // G*64
  unsigned* mx = (unsigned*)(sum + (size_t)G * 64);   // G*64
  int* counts  = (int*)(mx + (size_t)G * 64);         // G
  float* pooled = (float*)(counts + G);               // G*128
  float* hc     = pooled + (size_t)G * 128;           // G*64

  // ---- Layer 1 (K=12) ----
  hipMemsetAsync(agg, 0, (size_t)N * 64 * sizeof(float), stream);
  gine_scatter<12><<<cdiv((long long)E * 3, 256), 256, 0, stream>>>(x, ei, ea, We1, be1, agg, E);
  layer1_update<<<cdiv((long long)N * 64, 256), 256, 0, stream>>>(
      x, agg, W1, b1, g1, bt1, rm1, rv1, Wr, br, hA, N);

  // ---- Layer 2 (K=64, WMMA) ----
  hipMemsetAsync(agg, 0, (size_t)N * 64 * sizeof(float), stream);
  gine_scatter<64><<<cdiv((long long)E * 16, 256), 256, 0, stream>>>(hA, ei, ea, We2, be2, agg, E);
  gemm_n64_wmma<64, 4, true, true, true><<<cdiv(N, 64), 128, 0, stream>>>(
      hA, agg, W2, b2, g2, bt2, rm2, rv2, hA, hB, N);

  // ---- Layer 3 (K=64, WMMA) ----
  hipMemsetAsync(agg, 0, (size_t)N * 64 * sizeof(float), stream);
  gine_scatter<64><<<cdiv((long long)E * 16, 256), 256, 0, stream>>>(hB, ei, ea, We3, be3, agg, E);
  gemm_n64_wmma<64, 4, true, true, true><<<cdiv(N, 64), 128, 0, stream>>>(
      hB, agg, W3, b3, g3, bt3, rm3, rv3, hB, hA, N);

  // ---- Pooling ----
  pool_init<<<cdiv(G * 64, 256), 256, 0, stream>>>(sum, mx, counts, G);
  pool_scatter<<<cdiv((long long)N * 64, 256), 256, 0, stream>>>(hA, batch, sum, mx, counts, N);
  pool_finalize<<<cdiv(G * 64, 256), 256, 0, stream>>>(sum, mx, counts, pooled, G);

  // ---- Classifier: pooled[512x128] @ Wc1[128x64] (+relu) via WMMA ----
  gemm_n64_wmma<128, 2, false, false, false><<<cdiv(G, 32), 64, 0, stream>>>(
      pooled, nullptr, Wc1, bc1, nullptr, nullptr, nullptr, nullptr, nullptr, hc, G);
  // ---- Final 64 -> 10 ----
  cls_final<<<cdiv(G * 10, 256), 256, 0, stream>>>(hc, Wc2, bc2, (float*)d_out, G);
}